// TMC_9543417332439
// MI455X (gfx1250) — compile-verified
//
#include <hip/hip_runtime.h>

// out[row] = softmax_over_1000( d*r/1000 + d + r ),  d = x[row,1,:], r = x[row,0,:]
// (exact algebraic reduction of the DS-combination reference; softmax shift-invariance
//  removes the +1, and t in [0,2.002) makes max-subtraction unnecessary.)

typedef float v4f __attribute__((ext_vector_type(4)));
typedef float v2f __attribute__((ext_vector_type(2)));
typedef float v8f __attribute__((ext_vector_type(8)));

#define NCLS 1000
#define CHUNKS 250          // float4 chunks per class-row
#define ROWS_PER_BLOCK 8    // one wave32 per row, 256-thread blocks

// Full-precision f32 wave sum via two chained V_WMMA_F32_16X16X4_F32 — no LDS,
// no ds_bpermute, no cross-lane shuffle.
//
// Stage 1: A vgpr0 = p (lane partial), vgpr1 = 0 -> row m of A = [p_m, 0, p_{m+16}, 0]
//          B = all ones                           -> D1[m][n] = p_m + p_{m+16}
//          Lane sums its 8 D1 vgprs -> h = half-wave sums
//          (lanes 0-15: sum over m=0..7, lanes 16-31: sum over m=8..15)
// Stage 2: B vgpr0 = h, vgpr1 = 0; A = all ones.  Column n of B is sourced from
//          lanes n and n+16, so D2[m][n] = h_n + h_{n+16} = full 32-lane sum,
//          and by the C/D layout every lane's D2[0] holds that total.
__device__ __forceinline__ float wave_sum_f32(float p) {
    v2f ones; ones.x = 1.0f; ones.y = 1.0f;
    v8f C = {};

    v2f A1; A1.x = p; A1.y = 0.0f;
    v8f D1 = __builtin_amdgcn_wmma_f32_16x16x4_f32(
        false, A1, false, ones, (short)0, C, false, false);
    float h = ((D1[0] + D1[1]) + (D1[2] + D1[3])) + ((D1[4] + D1[5]) + (D1[6] + D1[7]));

    v2f B2; B2.x = h; B2.y = 0.0f;
    v8f D2 = __builtin_amdgcn_wmma_f32_16x16x4_f32(
        false, ones, false, B2, (short)0, C, false, false);
    return D2[0];
}

__global__ __launch_bounds__(256) void ds_combin_softmax_kernel(
    const float* __restrict__ x, float* __restrict__ out, int nrows) {
    const int wave = threadIdx.x >> 5;
    const int lane = threadIdx.x & 31;
    const int row  = blockIdx.x * ROWS_PER_BLOCK + wave;
    if (row >= nrows) return;  // wave-uniform; never taken with exact grid

    const v4f* __restrict__ rgb4 = (const v4f*)(x + (size_t)row * (2 * NCLS));
    const v4f* __restrict__ dep4 = (const v4f*)(x + (size_t)row * (2 * NCLS) + NCLS);
    v4f* __restrict__ out4       = (v4f*)(out + (size_t)row * NCLS);

    v4f  e[8];
    float part = 0.0f;

    // 250 float4 chunks / 32 lanes: 7 full iterations + 1 with lanes < 26 active.
    #pragma unroll
    for (int j = 0; j < 8; ++j) {
        const int c = lane + j * 32;
        if (j < 7 || lane < (CHUNKS - 7 * 32)) {
            v4f a = __builtin_nontemporal_load(&rgb4[c]);  // single-use: NT load
            v4f b = __builtin_nontemporal_load(&dep4[c]);
            v4f t;
            t.x = __fmaf_rn(a.x * b.x, 1.0e-3f, a.x + b.x);
            t.y = __fmaf_rn(a.y * b.y, 1.0e-3f, a.y + b.y);
            t.z = __fmaf_rn(a.z * b.z, 1.0e-3f, a.z + b.z);
            t.w = __fmaf_rn(a.w * b.w, 1.0e-3f, a.w + b.w);
            v4f ev;
            ev.x = __expf(t.x);   // v_exp_f32: quarter-rate, ~free vs 23.3 TB/s stream
            ev.y = __expf(t.y);
            ev.z = __expf(t.z);
            ev.w = __expf(t.w);
            e[j] = ev;
            part += (ev.x + ev.y) + (ev.z + ev.w);
        }
    }

    // EXEC is all-ones here (ragged tail has re-converged) -> WMMA legal.
    const float Z    = wave_sum_f32(part);
    const float invZ = 1.0f / Z;

    #pragma unroll
    for (int j = 0; j < 8; ++j) {
        const int c = lane + j * 32;
        if (j < 7 || lane < (CHUNKS - 7 * 32)) {
            __builtin_nontemporal_store(e[j] * invZ, &out4[c]);
        }
    }
}

extern "C" void kernel_launch(void* const* d_in, const int* in_sizes, int n_in,
                              void* d_out, int out_size, void* d_ws, size_t ws_size,
                              hipStream_t stream) {
    const float* x = (const float*)d_in[0];
    float* out = (float*)d_out;
    const int nrows = in_sizes[0] / (2 * NCLS);  // 65536
    const int grid  = (nrows + ROWS_PER_BLOCK - 1) / ROWS_PER_BLOCK;
    ds_combin_softmax_kernel<<<grid, 256, 0, stream>>>(x, out, nrows);
}